// RNN_2680059593011
// MI455X (gfx1250) — compile-verified
//
#include <hip/hip_runtime.h>

#define B_   64
#define T_   512
#define I_   128
#define H_   512
#define C_   5
#define BT_  (B_ * T_)
#define LDSW 516   // 512 + 4 floats pad: lane bank = (4*m + k) % 64 -> conflict-free

typedef float v2f __attribute__((ext_vector_type(2)));
typedef float v8f __attribute__((ext_vector_type(8)));

__device__ __forceinline__ v8f wmma_f32_k4(v2f a, v2f b, v8f c) {
    // D = A(16x4 f32) * B(4x16 f32) + C(16x16 f32)
    return __builtin_amdgcn_wmma_f32_16x16x4_f32(
        /*neg_a=*/false, a, /*neg_b=*/false, b,
        /*c_mod=*/(short)0, c, /*reuse_a=*/false, /*reuse_b=*/false);
}

__device__ __forceinline__ v2f ld2(const float* p) {
    return *(const v2f*)p;   // 8B-aligned by construction (even float offsets)
}

// ---------------------------------------------------------------------------
// Kernel A: xproj[n, h] = sum_d X[n, d] * W[h, d] + (b0[h] + b1[h])
// One wave per 16x16 output tile; A/B fragments per ISA 16x4 f32 layout:
//   lane l: row = l%16, kpair = (l/16)*2, vgpr{0,1} hold k, k+1.
// ---------------------------------------------------------------------------
template <int DIN>
__global__ __launch_bounds__(256) void xproj_gemm(const float* __restrict__ X,
                                                  const float* __restrict__ W,
                                                  const float* __restrict__ b0,
                                                  const float* __restrict__ b1,
                                                  float* __restrict__ out) {
    const int lane = threadIdx.x & 31;
    const int wid  = blockIdx.x * (blockDim.x >> 5) + (threadIdx.x >> 5);
    const int nt   = wid & (H_ / 16 - 1);   // 32 N tiles
    const int mt   = wid >> 5;              // 2048 M tiles
    const int mrow = mt * 16 + (lane & 15);
    const int nrow = nt * 16 + (lane & 15);
    const int kph  = (lane >> 4) * 2;

    const float* arow = X + (size_t)mrow * DIN + kph;
    const float* brow = W + (size_t)nrow * DIN + kph;  // B[k][n] = W[n][k]

    v8f c0 = {};
    v8f c1 = {};
#pragma unroll 4
    for (int k = 0; k < DIN; k += 8) {
        c0 = wmma_f32_k4(ld2(arow + k),     ld2(brow + k),     c0);
        c1 = wmma_f32_k4(ld2(arow + k + 4), ld2(brow + k + 4), c1);
    }
    v8f c = c0 + c1;

    const float bias  = b0[nrow] + b1[nrow];
    const int   mbase = mt * 16 + 8 * (lane >> 4);   // C layout: M = i + 8*(lane/16)
    const int   ncol  = nt * 16 + (lane & 15);
#pragma unroll
    for (int i = 0; i < 8; ++i)
        out[(size_t)(mbase + i) * H_ + ncol] = c[i] + bias;
}

// ---------------------------------------------------------------------------
// Kernel B: full-layer recurrent scan for one 16-row batch tile.
// h (16 x 512, f32) lives in LDS for all 512 steps. 32 waves = 32 N-tiles.
//   h_t = relu(xproj[:, t, :] + h_{t-1} @ W_hh^T)
// ---------------------------------------------------------------------------
__global__ __launch_bounds__(1024) void rnn_scan(const float* __restrict__ xproj,
                                                 const float* __restrict__ Whh,
                                                 float* __restrict__ ys) {
    __shared__ float hbuf[16 * LDSW];

    const int tid   = threadIdx.x;
    const int lane  = tid & 31;
    const int wave  = tid >> 5;        // 0..31 -> N tile
    const int btile = blockIdx.x;      // 0..3  -> 16 batch rows

    for (int i = tid; i < 16 * LDSW; i += 1024) hbuf[i] = 0.f;  // h0 = 0
    __syncthreads();

    const int nloc = lane & 15;
    const int ncol = wave * 16 + nloc;
    const int kph  = (lane >> 4) * 2;
    const int msub = 8 * (lane >> 4);

    const float* wr   = Whh + (size_t)ncol * H_ + kph;        // B frag: W_hh row ncol
    const float* arow = hbuf + (size_t)(lane & 15) * LDSW + kph;  // A frag: h row m

    for (int t = 0; t < T_; ++t) {
        v8f c0 = {};
        v8f c1 = {};
#pragma unroll 4
        for (int k = 0; k < H_; k += 8) {
            c0 = wmma_f32_k4(ld2(arow + k),     ld2(wr + k),     c0);
            c1 = wmma_f32_k4(ld2(arow + k + 4), ld2(wr + k + 4), c1);
        }
        v8f c = c0 + c1;

        const size_t gbase = ((size_t)(btile * 16 + msub) * T_ + t) * H_ + ncol;
#pragma unroll
        for (int i = 0; i < 8; ++i) {
            float v = c[i] + xproj[gbase + (size_t)i * (T_ * H_)];
            c[i] = v > 0.f ? v : 0.f;
        }

        __syncthreads();  // all waves finished reading h_{t-1} from LDS
#pragma unroll
        for (int i = 0; i < 8; ++i) {
            hbuf[(size_t)(msub + i) * LDSW + ncol] = c[i];
            ys[gbase + (size_t)i * (T_ * H_)] = c[i];
        }
        __syncthreads();  // h_t visible to all waves
    }
}

// ---------------------------------------------------------------------------
// Kernel C: head  out[b, c] = ys[b, T-1, :] . W_out[c, :] + b_out[c]
// ---------------------------------------------------------------------------
__global__ __launch_bounds__(320) void head_kernel(const float* __restrict__ ys,
                                                   const float* __restrict__ Wout,
                                                   const float* __restrict__ bout,
                                                   float* __restrict__ out) {
    int idx = threadIdx.x + blockIdx.x * blockDim.x;
    if (idx >= B_ * C_) return;
    int b = idx / C_;
    int c = idx % C_;
    const float* hr = ys + ((size_t)b * T_ + (T_ - 1)) * H_;
    const float* wc = Wout + (size_t)c * H_;
    float s = bout[c];
    for (int k = 0; k < H_; ++k) s += hr[k] * wc[k];
    out[idx] = s;
}

// ---------------------------------------------------------------------------
extern "C" void kernel_launch(void* const* d_in, const int* in_sizes, int n_in,
                              void* d_out, int out_size, void* d_ws, size_t ws_size,
                              hipStream_t stream) {
    const float* x       = (const float*)d_in[0];
    const float* Wih[3]  = {(const float*)d_in[1], (const float*)d_in[5], (const float*)d_in[9]};
    const float* Whh[3]  = {(const float*)d_in[2], (const float*)d_in[6], (const float*)d_in[10]};
    const float* bih[3]  = {(const float*)d_in[3], (const float*)d_in[7], (const float*)d_in[11]};
    const float* bhh[3]  = {(const float*)d_in[4], (const float*)d_in[8], (const float*)d_in[12]};
    const float* Wout    = (const float*)d_in[13];
    const float* bout    = (const float*)d_in[14];
    float*       outp    = (float*)d_out;

    float* xproj = (float*)d_ws;                 // BT_*H_ floats (64 MB)
    float* act   = xproj + (size_t)BT_ * H_;     // BT_*H_ floats (64 MB)

    const int gemmBlocks = (BT_ / 16) * (H_ / 16) / 8;  // 8192 blocks x 8 waves

    // Layer 0 (Din = I_), then layers 1,2 (Din = H_), ping-ponging xproj/act.
    xproj_gemm<I_><<<gemmBlocks, 256, 0, stream>>>(x, Wih[0], bih[0], bhh[0], xproj);
    rnn_scan<<<B_ / 16, 1024, 0, stream>>>(xproj, Whh[0], act);

    xproj_gemm<H_><<<gemmBlocks, 256, 0, stream>>>(act, Wih[1], bih[1], bhh[1], xproj);
    rnn_scan<<<B_ / 16, 1024, 0, stream>>>(xproj, Whh[1], act);

    xproj_gemm<H_><<<gemmBlocks, 256, 0, stream>>>(act, Wih[2], bih[2], bhh[2], xproj);
    rnn_scan<<<B_ / 16, 1024, 0, stream>>>(xproj, Whh[2], act);

    head_kernel<<<1, 320, 0, stream>>>(act, Wout, bout, outp);
}